// GCN_55688545960297
// MI455X (gfx1250) — compile-verified
//
#include <hip/hip_runtime.h>

typedef __attribute__((ext_vector_type(2))) float v2f;
typedef __attribute__((ext_vector_type(8))) float v8f;

#define N_ALIGN 50176  // 50000 rounded up to multiple of 1024 for buffer alignment

// ---------------- degree / normalization (computed once, reused 3 layers) ----

__global__ void deg_init_kernel(float* __restrict__ deg, int n) {
  int t = blockIdx.x * blockDim.x + threadIdx.x;
  if (t < n) deg[t] = 1.0f;  // self-loop contributes 1 to every node
}

__global__ void deg_acc_kernel(const int* __restrict__ dst, float* __restrict__ deg, int e) {
  int t = blockIdx.x * blockDim.x + threadIdx.x;
  if (t < e) atomicAdd(&deg[dst[t]], 1.0f);
}

__global__ void deg_rsqrt_kernel(float* __restrict__ deg, int n) {
  int t = blockIdx.x * blockDim.x + threadIdx.x;
  if (t < n) deg[t] = rsqrtf(deg[t]);  // deg >= 1 always
}

// ---------------- H[M x C] = X[M x 128] * W[128 x C] via V_WMMA_F32_16X16X4_F32

template <int C>
__global__ __launch_bounds__(128) void gemm_wmma_f32(const float* __restrict__ X,
                                                     const float* __restrict__ W,
                                                     float* __restrict__ H, int M) {
  __shared__ float wlds[128 * C];  // 64KB (C=128) / 32KB (C=64); WGP has 320KB
  for (int i = threadIdx.x; i < 128 * C / 4; i += blockDim.x)
    ((float4*)wlds)[i] = ((const float4*)W)[i];
  __syncthreads();

  const int lane = threadIdx.x & 31;
  const int wave = threadIdx.x >> 5;
  const int m0 = (blockIdx.x * 4 + wave) * 16;
  if (m0 >= M) return;  // wave-uniform guard; M % 16 == 0 so tiles never straddle

  const int r    = lane & 15;        // A row in tile / B,D column in tile
  const int koff = (lane >> 4) * 2;  // f32 WMMA A/B layout: VGPR0=K{0,2}, VGPR1=K{1,3}
  const int rhi  = (lane >> 4) * 8;  // D layout: upper lane half holds rows M+8..15

  // Preload the wave's full 16x128 A row-panel: 32 v2f fragments (64 VGPRs)
  const float* xrow = X + (size_t)(m0 + r) * 128;
  v2f afrag[32];
#pragma unroll
  for (int kk = 0; kk < 32; ++kk) {
    const int k = kk * 4 + koff;
    afrag[kk].x = xrow[k];
    afrag[kk].y = xrow[k + 1];
  }

#pragma unroll
  for (int nt = 0; nt < C / 16; ++nt) {
    const int n0 = nt * 16;
    v8f acc = {};
#pragma unroll
    for (int kk = 0; kk < 32; ++kk) {
      const int k = kk * 4 + koff;
      v2f b;
      b.x = wlds[k * C + n0 + r];
      b.y = wlds[(k + 1) * C + n0 + r];
      acc = __builtin_amdgcn_wmma_f32_16x16x4_f32(false, afrag[kk], false, b,
                                                  (short)0, acc, false, false);
    }
#pragma unroll
    for (int g = 0; g < 8; ++g)
      H[(size_t)(m0 + rhi + g) * C + n0 + r] = acc[g];
  }
}

// ---------------- self-loop term initializes the accumulator ------------------

template <int C>
__global__ void self_loop_kernel(const float* __restrict__ H, const float* __restrict__ dinv,
                                 float* __restrict__ out, int n) {
  int t = blockIdx.x * blockDim.x + threadIdx.x;
  if (t < n * C) {
    float di = dinv[t / C];  // C is a power of two -> shift
    out[t] = H[t] * di * di;
  }
}

// ---------------- edge scatter: one wave per edge, L2-resident atomics --------

template <int C>
__global__ void scatter_kernel(const float* __restrict__ H, const int* __restrict__ src,
                               const int* __restrict__ dst, const float* __restrict__ dinv,
                               float* __restrict__ out, int e) {
  const int lane = threadIdx.x & 31;
  const int w = (blockIdx.x * blockDim.x + threadIdx.x) >> 5;
  if (w >= e) return;
  const int s = src[w], d = dst[w];
  const float nrm = dinv[s] * dinv[d];
  const float* hrow = H + (size_t)s * C;
  float* orow = out + (size_t)d * C;
  if constexpr (C == 128) {
    float4 v = ((const float4*)hrow)[lane];  // global_load_b128, coalesced
    const int c = lane * 4;
    atomicAdd(&orow[c + 0], v.x * nrm);
    atomicAdd(&orow[c + 1], v.y * nrm);
    atomicAdd(&orow[c + 2], v.z * nrm);
    atomicAdd(&orow[c + 3], v.w * nrm);
  } else {
    float2 v = ((const float2*)hrow)[lane];
    const int c = lane * 2;
    atomicAdd(&orow[c + 0], v.x * nrm);
    atomicAdd(&orow[c + 1], v.y * nrm);
  }
}

// ---------------- bias (+ optional ReLU) epilogue -----------------------------

template <int C, bool RELU>
__global__ void bias_act_kernel(float* __restrict__ out, const float* __restrict__ b, int n) {
  int t = blockIdx.x * blockDim.x + threadIdx.x;
  if (t < n * C) {
    float v = out[t] + b[t & (C - 1)];
    out[t] = RELU ? fmaxf(v, 0.0f) : v;
  }
}

// ---------------- host orchestration ------------------------------------------

extern "C" void kernel_launch(void* const* d_in, const int* in_sizes, int n_in,
                              void* d_out, int out_size, void* d_ws, size_t ws_size,
                              hipStream_t stream) {
  const float* x  = (const float*)d_in[0];
  const int*   ei = (const int*)d_in[1];  // [2, E]: row 0 = src, row 1 = dst
  const float* W1 = (const float*)d_in[2];
  const float* b1 = (const float*)d_in[3];
  const float* W2 = (const float*)d_in[4];
  const float* b2 = (const float*)d_in[5];
  const float* W3 = (const float*)d_in[6];
  const float* b3 = (const float*)d_in[7];

  const int N = in_sizes[0] / 128;
  const int E = in_sizes[1] / 2;
  const int* src = ei;
  const int* dst = ei + E;

  float* dinv = (float*)d_ws;                          // N floats
  float* Hbuf = dinv + N_ALIGN;                        // N x 128
  float* bufA = Hbuf + (size_t)N_ALIGN * 128;          // N x 128
  float* bufB = bufA + (size_t)N_ALIGN * 128;          // N x 128
  float* outp = (float*)d_out;                         // N x 64

  const int T = 256;
  const int nBlk   = (N + T - 1) / T;
  const int eBlk   = (E + T - 1) / T;
  const int gBlk   = (N + 63) / 64;                    // 4 waves x 16 rows per block
  const int sBlk   = (E + (T >> 5) - 1) / (T >> 5);    // one wave per edge
  const int e128   = (N * 128 + T - 1) / T;
  const int e64    = (N * 64 + T - 1) / T;

  // normalization (shared by all layers)
  deg_init_kernel<<<nBlk, T, 0, stream>>>(dinv, N);
  deg_acc_kernel<<<eBlk, T, 0, stream>>>(dst, dinv, E);
  deg_rsqrt_kernel<<<nBlk, T, 0, stream>>>(dinv, N);

  // layer 1: 128 -> 128, ReLU
  gemm_wmma_f32<128><<<gBlk, 128, 0, stream>>>(x, W1, Hbuf, N);
  self_loop_kernel<128><<<e128, T, 0, stream>>>(Hbuf, dinv, bufA, N);
  scatter_kernel<128><<<sBlk, T, 0, stream>>>(Hbuf, src, dst, dinv, bufA, E);
  bias_act_kernel<128, true><<<e128, T, 0, stream>>>(bufA, b1, N);

  // layer 2: 128 -> 128, ReLU
  gemm_wmma_f32<128><<<gBlk, 128, 0, stream>>>(bufA, W2, Hbuf, N);
  self_loop_kernel<128><<<e128, T, 0, stream>>>(Hbuf, dinv, bufB, N);
  scatter_kernel<128><<<sBlk, T, 0, stream>>>(Hbuf, src, dst, dinv, bufB, E);
  bias_act_kernel<128, true><<<e128, T, 0, stream>>>(bufB, b2, N);

  // layer 3: 128 -> 64, no ReLU, writes d_out
  gemm_wmma_f32<64><<<gBlk, 128, 0, stream>>>(bufB, W3, Hbuf, N);
  self_loop_kernel<64><<<e64, T, 0, stream>>>(Hbuf, dinv, outp, N);
  scatter_kernel<64><<<sBlk, T, 0, stream>>>(Hbuf, src, dst, dinv, outp, E);
  bias_act_kernel<64, false><<<e64, T, 0, stream>>>(outp, b3, N);
}